// TAHIN_52458730553653
// MI455X (gfx1250) — compile-verified
//
#include <hip/hip_runtime.h>

typedef float v2f __attribute__((ext_vector_type(2)));
typedef float v8f __attribute__((ext_vector_type(8)));
typedef unsigned u32x4 __attribute__((ext_vector_type(4)));
typedef int i32x4 __attribute__((ext_vector_type(4)));
typedef int i32x8 __attribute__((ext_vector_type(8)));

#define NN 50000       // nodes per side (N_U == N_I)
#define DD 64          // feature dim
#define HH 128         // semantic attention hidden
#define EE 500000      // edges per graph
#define NEG_SLOPE 0.01f

// ---------- monotone float<->uint order encoding for atomic max ----------
static __device__ __forceinline__ unsigned enc_ord(float f) {
    unsigned u = __float_as_uint(f);
    return (u & 0x80000000u) ? ~u : (u | 0x80000000u);
}
static __device__ __forceinline__ float dec_ord(unsigned u) {
    unsigned b = (u & 0x80000000u) ? (u ^ 0x80000000u) : ~u;
    return __uint_as_float(b);
}
static __device__ __forceinline__ float elu_f(float x) {
    return x > 0.f ? x : (__expf(x) - 1.f);
}

// ---------- Tensor Data Mover: 2D f32 weight tile (dim1 rows x dim0 cols,
// contiguous row-major) from global to LDS. D# layout per CDNA5 ISA ch.8. ----------
static __device__ __forceinline__ void tdm_load_2d_f32(const void* gptr, unsigned lds_off,
                                                       unsigned dim0, unsigned dim1) {
    unsigned long long ga = (unsigned long long)gptr;
    u32x4 g0;
    g0[0] = 1u;                                   // count=1, user descriptor
    g0[1] = lds_off;                              // lds_addr (bytes)
    g0[2] = (unsigned)ga;                         // global_addr[31:0]
    g0[3] = ((unsigned)(ga >> 32) & 0x01FFFFFFu)  // global_addr[56:32]
            | 0x80000000u;                        // type=2 ("image")
    i32x8 g1;
    g1[0] = 0x00020000;                           // data_size=2 (4 bytes), no multicast
    g1[1] = (int)(dim0 << 16);                    // tensor_dim0 lo16 at bits[63:48]
    g1[2] = (int)((dim0 >> 16) | (dim1 << 16));   // tensor_dim0 hi, tensor_dim1 lo
    g1[3] = (int)((dim1 >> 16) | (dim0 << 16));   // tensor_dim1 hi, tile_dim0 = dim0
    g1[4] = (int)dim1;                            // tile_dim1 = dim1, tile_dim2 = 0
    g1[5] = (int)dim0;                            // tensor_dim0_stride = dim0
    g1[6] = 0;
    g1[7] = 0;
    i32x4 gz = {0, 0, 0, 0};
#if __clang_major__ >= 23
    i32x8 gz8 = {0, 0, 0, 0, 0, 0, 0, 0};
    __builtin_amdgcn_tensor_load_to_lds(g0, g1, gz, gz, gz8, 0);
#else
    __builtin_amdgcn_tensor_load_to_lds(g0, g1, gz, gz, 0);
#endif
}

// ---------- utility kernels ----------
__global__ void tahin_zero(float* p, long n) {
    long i = (long)blockIdx.x * blockDim.x + threadIdx.x;
    long stride = (long)gridDim.x * blockDim.x;
    for (; i < n; i += stride) p[i] = 0.f;
}
__global__ void tahin_initmax(unsigned* p, int n) {
    int i = blockIdx.x * blockDim.x + threadIdx.x;
    if (i < n) p[i] = 0x007FFFFFu;  // enc(-inf)
}
// per-node dot(feat[i,:64], a[:64])
__global__ void tahin_nodedot(const float* __restrict__ feat, const float* __restrict__ a,
                              float* __restrict__ out, int n) {
    int i = blockIdx.x * blockDim.x + threadIdx.x;
    if (i >= n) return;
    const float4* f = (const float4*)(feat + (long)i * DD);
    const float4* av = (const float4*)a;
    float s = 0.f;
#pragma unroll
    for (int j = 0; j < DD / 4; ++j) {
        float4 x = f[j], y = av[j];
        s += x.x * y.x + x.y * y.y + x.z * y.z + x.w * y.w;
    }
    out[i] = s;
}
// ---------- GAT edge softmax ----------
__global__ void tahin_edge_max(const int* __restrict__ src, const int* __restrict__ dst,
                               const float* __restrict__ el, const float* __restrict__ er,
                               float* __restrict__ ebuf, unsigned* __restrict__ mmax, int ne) {
    int i = blockIdx.x * blockDim.x + threadIdx.x;
    if (i >= ne) return;
    float e = el[src[i]] + er[dst[i]];
    e = e > 0.f ? e : NEG_SLOPE * e;
    ebuf[i] = e;
    atomicMax(&mmax[dst[i]], enc_ord(e));
}
__global__ void tahin_edge_expsum(const int* __restrict__ dst, float* __restrict__ ebuf,
                                  const unsigned* __restrict__ mmax, float* __restrict__ ssum, int ne) {
    int i = blockIdx.x * blockDim.x + threadIdx.x;
    if (i >= ne) return;
    int d = dst[i];
    float ex = __expf(ebuf[i] - dec_ord(mmax[d]));
    ebuf[i] = ex;
    atomicAdd(&ssum[d], ex);
}
// wave-per-edge weighted scatter: zout[dst, moff + 0..63] += a * feat_src[src, :]
__global__ void tahin_edge_scatter(const int* __restrict__ src, const int* __restrict__ dst,
                                   const float* __restrict__ feat_src, const float* __restrict__ ebuf,
                                   const float* __restrict__ ssum, float* __restrict__ zout,
                                   int moff, int ne) {
    int wave = (blockIdx.x * blockDim.x + threadIdx.x) >> 5;
    int lane = threadIdx.x & 31;
    if (wave >= ne) return;
    int s = src[wave], d = dst[wave];
    float a = ebuf[wave] / ssum[d];
    const float* fs = feat_src + (long)s * DD;
    float* zo = zout + (long)d * (2 * DD) + moff;
    atomicAdd(&zo[lane], fs[lane] * a);
    atomicAdd(&zo[lane + 32], fs[lane + 32] * a);
}
__global__ void tahin_elu(float* p, long n) {
    long i = (long)blockIdx.x * blockDim.x + threadIdx.x;
    if (i < n) p[i] = elu_f(p[i]);
}
// ---------- GraphConv ----------
__global__ void tahin_degree(const int* __restrict__ idx, float* __restrict__ deg, int ne) {
    int i = blockIdx.x * blockDim.x + threadIdx.x;
    if (i < ne) atomicAdd(&deg[idx[i]], 1.0f);
}
__global__ void tahin_invsqrt(float* deg, int n) {
    int i = blockIdx.x * blockDim.x + threadIdx.x;
    if (i < n) { float d = deg[i]; deg[i] = rsqrtf(d > 0.f ? d : 1.0f); }
}
__global__ void tahin_gc_scatter(const int* __restrict__ src, const int* __restrict__ dst,
                                 const float* __restrict__ feat, const float* __restrict__ no,
                                 float* __restrict__ agg, int ne) {
    int wave = (blockIdx.x * blockDim.x + threadIdx.x) >> 5;
    int lane = threadIdx.x & 31;
    if (wave >= ne) return;
    int s = src[wave], d = dst[wave];
    float c = no[s];
    const float* fs = feat + (long)s * DD;
    float* ag = agg + (long)d * DD;
    atomicAdd(&ag[lane], fs[lane] * c);
    atomicAdd(&ag[lane + 32], fs[lane + 32] * c);
}
// h[row, moff + col] = elu( (ni[row]*agg[row,:]) @ W + b ), via fp32 WMMA 16x16x4.
// W (64x64, 16KB) staged into LDS by the Tensor Data Mover.
// One wave per 16-row tile; K=64 in 16 steps; 4 column tiles. n % 16 == 0.
__global__ __launch_bounds__(128)
void tahin_gc_matmul(const float* __restrict__ agg, const float* __restrict__ ni,
                     const float* __restrict__ W, const float* __restrict__ b,
                     float* __restrict__ hout, int moff, int n) {
    __shared__ float Ws[DD * DD];
    if (threadIdx.x < 32) {
        tdm_load_2d_f32(W, (unsigned)(unsigned long long)(void*)Ws, DD, DD);
        __builtin_amdgcn_s_wait_tensorcnt(0);
    }
    __syncthreads();
    int wave = threadIdx.x >> 5;
    int lane = threadIdx.x & 31;
    int tile = blockIdx.x * 4 + wave;
    int ntiles = n >> 4;
    if (tile >= ntiles) return;
    int rb = tile * 16;
    int r0 = rb + (lane & 15);
    int khalf = (lane >> 4) * 2;        // lanes 0-15 -> K 0,1 ; lanes 16-31 -> K 2,3
    float scale = ni[r0];
    v8f acc[4] = {};
    for (int kt = 0; kt < 16; ++kt) {
        int k0 = kt * 4 + khalf;
        v2f a;
        a.x = agg[(long)r0 * DD + k0] * scale;
        a.y = agg[(long)r0 * DD + k0 + 1] * scale;
#pragma unroll
        for (int ct = 0; ct < 4; ++ct) {
            v2f bb;
            bb.x = Ws[k0 * DD + ct * 16 + (lane & 15)];
            bb.y = Ws[(k0 + 1) * DD + ct * 16 + (lane & 15)];
            acc[ct] = __builtin_amdgcn_wmma_f32_16x16x4_f32(
                false, a, false, bb, (short)0, acc[ct], false, false);
        }
    }
#pragma unroll
    for (int ct = 0; ct < 4; ++ct) {
        int col = ct * 16 + (lane & 15);
        float bias = b[col];
#pragma unroll
        for (int r = 0; r < 8; ++r) {
            int row = rb + r + ((lane >> 4) << 3);
            float v = acc[ct][r] + bias;
            hout[(long)row * (2 * DD) + moff + col] = elu_f(v);
        }
    }
}
// ---------- semantic attention score: accum += mean_n( tanh(z[n,m,:]@W1 + b1) @ w2 ) ----------
// W1 (64x128, 32KB) staged into LDS by the Tensor Data Mover.
__global__ __launch_bounds__(128)
void tahin_sem_score(const float* __restrict__ z, int moff,
                     const float* __restrict__ W1, const float* __restrict__ b1,
                     const float* __restrict__ w2, float* __restrict__ score, int n) {
    __shared__ float W1s[DD * HH];
    __shared__ float rowsum[4][16];
    if (threadIdx.x < 32) {
        tdm_load_2d_f32(W1, (unsigned)(unsigned long long)(void*)W1s, HH, DD);
        __builtin_amdgcn_s_wait_tensorcnt(0);
    }
    __syncthreads();
    int wave = threadIdx.x >> 5;
    int lane = threadIdx.x & 31;
    int tile = blockIdx.x * 4 + wave;
    int ntiles = n >> 4;
    if (tile >= ntiles) return;
    if (lane < 16) rowsum[wave][lane] = 0.f;   // same-wave LDS ops stay in order
    int rb = tile * 16;
    int r0 = rb + (lane & 15);
    int khalf = (lane >> 4) * 2;
    for (int ct = 0; ct < 8; ++ct) {           // HH/16 column tiles
        v8f acc = {};
        for (int kt = 0; kt < 16; ++kt) {
            int k0 = kt * 4 + khalf;
            v2f a;
            a.x = z[(long)r0 * (2 * DD) + moff + k0];
            a.y = z[(long)r0 * (2 * DD) + moff + k0 + 1];
            v2f bb;
            bb.x = W1s[k0 * HH + ct * 16 + (lane & 15)];
            bb.y = W1s[(k0 + 1) * HH + ct * 16 + (lane & 15)];
            acc = __builtin_amdgcn_wmma_f32_16x16x4_f32(
                false, a, false, bb, (short)0, acc, false, false);
        }
        int col = ct * 16 + (lane & 15);
        float b1c = b1[col], w2c = w2[col];
#pragma unroll
        for (int r = 0; r < 8; ++r) {
            int rloc = r + ((lane >> 4) << 3);
            atomicAdd(&rowsum[wave][rloc], tanhf(acc[r] + b1c) * w2c);
        }
    }
    if (lane < 16) atomicAdd(score, rowsum[wave][lane] * (1.0f / (float)n));
}
// ---------- final combine: out = softmax2(rel scores) . z + softmax2(han scores) . h ----------
__global__ void tahin_combine(const float* __restrict__ zu, const float* __restrict__ zi,
                              const float* __restrict__ hu, const float* __restrict__ hi,
                              const float* __restrict__ score, float* __restrict__ out) {
    long i = (long)blockIdx.x * blockDim.x + threadIdx.x;
    long total = (long)(2 * NN) * DD;
    if (i >= total) return;
    long node = i >> 6;
    int d = (int)(i & 63);
    const float* z;
    const float* h;
    float s0, s1, t0, t1;
    if (node < NN) {
        z = zu + node * (2 * DD); h = hu + node * (2 * DD);
        s0 = score[0]; s1 = score[1]; t0 = score[4]; t1 = score[5];
    } else {
        long m = node - NN;
        z = zi + m * (2 * DD); h = hi + m * (2 * DD);
        s0 = score[2]; s1 = score[3]; t0 = score[6]; t1 = score[7];
    }
    float ms = fmaxf(s0, s1), e0 = __expf(s0 - ms), e1 = __expf(s1 - ms);
    float br0 = e0 / (e0 + e1), br1 = 1.f - br0;
    float mt = fmaxf(t0, t1), f0 = __expf(t0 - mt), f1 = __expf(t1 - mt);
    float bh0 = f0 / (f0 + f1), bh1 = 1.f - bh0;
    out[i] = br0 * z[d] + br1 * z[DD + d] + bh0 * h[d] + bh1 * h[DD + d];
}

// ---------- host-side launch ----------
extern "C" void kernel_launch(void* const* d_in, const int* in_sizes, int n_in,
                              void* d_out, int out_size, void* d_ws, size_t ws_size,
                              hipStream_t stream) {
    const float* feat_user = (const float*)d_in[0];
    const float* feat_item = (const float*)d_in[1];
    const float* attn_l    = (const float*)d_in[2];
    const float* attn_r    = (const float*)d_in[3];
    const float* W_gc      = (const float*)d_in[4];
    const float* b_gc      = (const float*)d_in[5];
    const float* sa_rel_W1 = (const float*)d_in[6];
    const float* sa_rel_b1 = (const float*)d_in[7];
    const float* sa_rel_w2 = (const float*)d_in[8];
    const float* sa_u_W1   = (const float*)d_in[9];
    const float* sa_u_b1   = (const float*)d_in[10];
    const float* sa_u_w2   = (const float*)d_in[11];
    const float* sa_i_W1   = (const float*)d_in[12];
    const float* sa_i_b1   = (const float*)d_in[13];
    const float* sa_i_w2   = (const float*)d_in[14];
    const int* rel_u_src = (const int*)d_in[15];
    const int* rel_u_dst = (const int*)d_in[16];
    const int* rel_i_src = (const int*)d_in[17];
    const int* rel_i_dst = (const int*)d_in[18];
    const int* mp_u_src  = (const int*)d_in[19];
    const int* mp_u_dst  = (const int*)d_in[20];
    const int* mp_i_src  = (const int*)d_in[21];
    const int* mp_i_dst  = (const int*)d_in[22];

    float* ws = (float*)d_ws;
    const long ZSZ = (long)NN * 2 * DD;          // 6,400,000
    float* zu   = ws;
    float* zi   = zu + ZSZ;
    float* hu   = zi + ZSZ;
    float* hi   = hu + ZSZ;
    float* ebuf = hi + ZSZ;                      // EE
    unsigned* mmax = (unsigned*)(ebuf + EE);     // NN
    float* ssum = (float*)(mmax + NN);           // NN
    float* el   = ssum + NN;                     // NN
    float* er   = el + NN;                       // NN
    float* dego = er + NN;                       // NN
    float* degi = dego + NN;                     // NN
    float* agg  = degi + NN;                     // NN*DD = 3,200,000
    float* score = agg + (long)NN * DD;          // 8

    const int T = 256;
    const int gN   = (NN + T - 1) / T;
    const int gE   = (EE + T - 1) / T;
    const int gEw  = (EE * 32 + T - 1) / T;      // wave-per-edge
    const int gZ   = (int)((ZSZ + T - 1) / T);
    const int gMM  = ((NN >> 4) + 3) / 4;        // 4 waves/block over 16-row tiles

    // zero accumulated buffers
    tahin_zero<<<gZ, T, 0, stream>>>(zu, ZSZ);
    tahin_zero<<<gZ, T, 0, stream>>>(zi, ZSZ);
    tahin_zero<<<1, 32, 0, stream>>>(score, 8);

    // ---- 4 GAT relation graphs ----
    const int*   g_src[4] = { rel_u_src, rel_u_src + EE, rel_i_src, rel_i_src + EE };
    const int*   g_dst[4] = { rel_u_dst, rel_u_dst + EE, rel_i_dst, rel_i_dst + EE };
    const float* g_fs [4] = { feat_item, feat_item, feat_user, feat_user };
    const float* g_fd [4] = { feat_user, feat_user, feat_item, feat_item };
    float*       g_z  [4] = { zu, zu, zi, zi };
    const int    g_mo [4] = { 0, DD, 0, DD };
    for (int r = 0; r < 4; ++r) {
        tahin_nodedot<<<gN, T, 0, stream>>>(g_fs[r], attn_l + r * DD, el, NN);
        tahin_nodedot<<<gN, T, 0, stream>>>(g_fd[r], attn_r + r * DD, er, NN);
        tahin_initmax<<<gN, T, 0, stream>>>(mmax, NN);
        tahin_zero<<<gN, T, 0, stream>>>(ssum, NN);
        tahin_edge_max<<<gE, T, 0, stream>>>(g_src[r], g_dst[r], el, er, ebuf, mmax, EE);
        tahin_edge_expsum<<<gE, T, 0, stream>>>(g_dst[r], ebuf, mmax, ssum, EE);
        tahin_edge_scatter<<<gEw, T, 0, stream>>>(g_src[r], g_dst[r], g_fs[r], ebuf, ssum,
                                                  g_z[r], g_mo[r], EE);
    }
    tahin_elu<<<gZ, T, 0, stream>>>(zu, ZSZ);
    tahin_elu<<<gZ, T, 0, stream>>>(zi, ZSZ);

    // ---- 4 GraphConv meta-path graphs (WMMA matmul, TDM-staged weights) ----
    const int*   c_src[4] = { mp_u_src, mp_u_src + EE, mp_i_src, mp_i_src + EE };
    const int*   c_dst[4] = { mp_u_dst, mp_u_dst + EE, mp_i_dst, mp_i_dst + EE };
    const float* c_f  [4] = { feat_user, feat_user, feat_item, feat_item };
    float*       c_h  [4] = { hu, hu, hi, hi };
    const int    c_mo [4] = { 0, DD, 0, DD };
    for (int g = 0; g < 4; ++g) {
        tahin_zero<<<gN, T, 0, stream>>>(dego, NN);
        tahin_zero<<<gN, T, 0, stream>>>(degi, NN);
        tahin_zero<<<(int)(((long)NN * DD + T - 1) / T), T, 0, stream>>>(agg, (long)NN * DD);
        tahin_degree<<<gE, T, 0, stream>>>(c_src[g], dego, EE);
        tahin_degree<<<gE, T, 0, stream>>>(c_dst[g], degi, EE);
        tahin_invsqrt<<<gN, T, 0, stream>>>(dego, NN);
        tahin_invsqrt<<<gN, T, 0, stream>>>(degi, NN);
        tahin_gc_scatter<<<gEw, T, 0, stream>>>(c_src[g], c_dst[g], c_f[g], dego, agg, EE);
        tahin_gc_matmul<<<gMM, 128, 0, stream>>>(agg, degi, W_gc + (long)g * DD * DD,
                                                 b_gc + g * DD, c_h[g], c_mo[g], NN);
    }

    // ---- semantic attention scores (WMMA, TDM-staged W1) ----
    tahin_sem_score<<<gMM, 128, 0, stream>>>(zu, 0,  sa_rel_W1, sa_rel_b1, sa_rel_w2, score + 0, NN);
    tahin_sem_score<<<gMM, 128, 0, stream>>>(zu, DD, sa_rel_W1, sa_rel_b1, sa_rel_w2, score + 1, NN);
    tahin_sem_score<<<gMM, 128, 0, stream>>>(zi, 0,  sa_rel_W1, sa_rel_b1, sa_rel_w2, score + 2, NN);
    tahin_sem_score<<<gMM, 128, 0, stream>>>(zi, DD, sa_rel_W1, sa_rel_b1, sa_rel_w2, score + 3, NN);
    tahin_sem_score<<<gMM, 128, 0, stream>>>(hu, 0,  sa_u_W1, sa_u_b1, sa_u_w2, score + 4, NN);
    tahin_sem_score<<<gMM, 128, 0, stream>>>(hu, DD, sa_u_W1, sa_u_b1, sa_u_w2, score + 5, NN);
    tahin_sem_score<<<gMM, 128, 0, stream>>>(hi, 0,  sa_i_W1, sa_i_b1, sa_i_w2, score + 6, NN);
    tahin_sem_score<<<gMM, 128, 0, stream>>>(hi, DD, sa_i_W1, sa_i_b1, sa_i_w2, score + 7, NN);

    // ---- combine ----
    long total = (long)(2 * NN) * DD;
    tahin_combine<<<(int)((total + T - 1) / T), T, 0, stream>>>(zu, zi, hu, hi, score,
                                                                (float*)d_out);
}